// CourseRecommender_46205258170382
// MI455X (gfx1250) — compile-verified
//
#include <hip/hip_runtime.h>
#include <cstdint>

typedef float v2f __attribute__((ext_vector_type(2)));
typedef float v8f __attribute__((ext_vector_type(8)));

#define HDIM 64
#define NEG_SLOPE 0.2f

// ---- order-preserving float<->int encoding for atomicMax-based segment max ----
__device__ __forceinline__ int enc_f32(float f) {
  int i = __float_as_int(f);
  return i >= 0 ? i : (i ^ 0x7fffffff);
}
__device__ __forceinline__ float dec_f32(int i) {
  int j = i >= 0 ? i : (i ^ 0x7fffffff);
  return __int_as_float(j);
}

// ---------------------------------------------------------------------------
// h = x @ W  via V_WMMA_F32_16X16X4_F32. One wave computes a 16-row x 64-col
// slab of h (4 accumulators of 16x16), K-loop over 64 in steps of 4.
// A layout (ISA 7.12.2, 32-bit A 16x4): lane t -> row t&15, a.x=K(kk), a.y=K(kk+1),
// kk = (t>>4)*2. B mirrored. C/D: vgpr r -> row r+(t>>4)*8, col t&15.
// ---------------------------------------------------------------------------
__global__ void gat_proj_wmma(const float* __restrict__ x,
                              const float* __restrict__ W,
                              float* __restrict__ h, int nrows) {
  __shared__ float lW[HDIM * HDIM];
  for (int i = threadIdx.x; i < HDIM * HDIM; i += blockDim.x) lW[i] = W[i];
  __syncthreads();

  const int lane = threadIdx.x & 31;
  const int wave = threadIdx.x >> 5;
  const int rb   = blockIdx.x * (blockDim.x >> 5) + wave;   // 16-row block id
  if (rb * 16 >= nrows) return;

  const int m  = lane & 15;
  const int kk = (lane >> 4) << 1;            // 0 or 2
  const long row = (long)rb * 16 + m;
  const float* xr = x + row * HDIM;

  v8f acc0 = {}, acc1 = {}, acc2 = {}, acc3 = {};
#pragma unroll
  for (int kb = 0; kb < HDIM; kb += 4) {
    v2f a;
    a.x = xr[kb + kk];
    a.y = xr[kb + kk + 1];
    const float* w0 = &lW[(kb + kk) * HDIM];
    const float* w1 = &lW[(kb + kk + 1) * HDIM];
    v2f b0, b1, b2, b3;
    b0.x = w0[m];      b0.y = w1[m];
    b1.x = w0[16 + m]; b1.y = w1[16 + m];
    b2.x = w0[32 + m]; b2.y = w1[32 + m];
    b3.x = w0[48 + m]; b3.y = w1[48 + m];
    acc0 = __builtin_amdgcn_wmma_f32_16x16x4_f32(false, a, false, b0, (short)0, acc0, false, false);
    acc1 = __builtin_amdgcn_wmma_f32_16x16x4_f32(false, a, false, b1, (short)0, acc1, false, false);
    acc2 = __builtin_amdgcn_wmma_f32_16x16x4_f32(false, a, false, b2, (short)0, acc2, false, false);
    acc3 = __builtin_amdgcn_wmma_f32_16x16x4_f32(false, a, false, b3, (short)0, acc3, false, false);
  }

  const int rowoff = (lane >> 4) * 8;
#pragma unroll
  for (int r = 0; r < 8; ++r) {
    long orow = (long)rb * 16 + rowoff + r;
    float* hp = h + orow * HDIM + m;
    hp[0]  = acc0[r];
    hp[16] = acc1[r];
    hp[32] = acc2[r];
    hp[48] = acc3[r];
  }
}

// ---- per-node scores: one wave32 per node, shuffle reduction -----------------
__global__ void gat_scores(const float* __restrict__ h,
                           const float* __restrict__ a_src,
                           const float* __restrict__ a_dst,
                           float* __restrict__ s, float* __restrict__ d,
                           int nnodes) {
  const int lane = threadIdx.x & 31;
  const int node = (int)(((long)blockIdx.x * blockDim.x + threadIdx.x) >> 5);
  if (node >= nnodes) return;
  const float* hr = h + (long)node * HDIM;
  float h0 = hr[lane], h1 = hr[32 + lane];
  float ss = h0 * a_src[lane] + h1 * a_src[32 + lane];
  float dd = h0 * a_dst[lane] + h1 * a_dst[32 + lane];
#pragma unroll
  for (int off = 16; off > 0; off >>= 1) {
    ss += __shfl_xor(ss, off, 32);
    dd += __shfl_xor(dd, off, 32);
  }
  if (lane == 0) { s[node] = ss; d[node] = dd; }
}

// ---- init accumulators -------------------------------------------------------
__global__ void gat_init(float* __restrict__ acc, float* __restrict__ z,
                         int* __restrict__ mx, int total) {
  const int idx = blockIdx.x * blockDim.x + threadIdx.x;
  if (idx >= total) return;
  acc[idx] = 0.f;
  if ((idx & (HDIM - 1)) == 0) {
    z[idx >> 6]  = 0.f;
    mx[idx >> 6] = (int)0x80000000;   // encoded -inf
  }
}

// ---- pass 1 over edges: segment max of leaky_relu(s[src]+d[dst]) ------------
__global__ void gat_edge_max(const int* __restrict__ ei, int E, int nnodes,
                             const float* __restrict__ s, const float* __restrict__ d,
                             int* __restrict__ mx) {
  const int tid = blockIdx.x * blockDim.x + threadIdx.x;
  const int EN = E + nnodes;
  if (tid >= EN) return;
  int src, dst;
  if (tid < E) { src = ei[tid]; dst = ei[E + tid]; }
  else         { src = dst = tid - E; }
  float e = s[src] + d[dst];
  e = e > 0.f ? e : NEG_SLOPE * e;
  atomicMax(&mx[dst], enc_f32(e));
}

// ---- pass 2 over edges: one wave per edge; acc[dst] += w*h[src], z[dst] += w -
__global__ void gat_edge_accum(const int* __restrict__ ei, int E, int nnodes,
                               const float* __restrict__ s, const float* __restrict__ d,
                               const int* __restrict__ mx, const float* __restrict__ h,
                               float* __restrict__ z, float* __restrict__ acc) {
  const long tid = (long)blockIdx.x * blockDim.x + threadIdx.x;
  const int  lane = (int)(tid & 31);
  const long eid  = tid >> 5;
  const long EN   = (long)E + nnodes;
  if (eid >= EN) return;
  int src, dst;
  if (eid < E) { src = ei[eid]; dst = ei[E + eid]; }
  else         { src = dst = (int)(eid - E); }
  float e = s[src] + d[dst];
  e = e > 0.f ? e : NEG_SLOPE * e;
  const float w = expf(e - dec_f32(mx[dst]));
  if (lane == 0) atomicAdd(&z[dst], w);
  const float* hr = h + (long)src * HDIM;
  float* ar = acc + (long)dst * HDIM;
  atomicAdd(&ar[lane],      w * hr[lane]);
  atomicAdd(&ar[32 + lane], w * hr[32 + lane]);
}

// ---- finalize: out = acc / z + b, optional ReLU ------------------------------
__global__ void gat_finalize(const float* __restrict__ acc, const float* __restrict__ z,
                             const float* __restrict__ b, float* __restrict__ out,
                             int total, int do_relu) {
  const int idx = blockIdx.x * blockDim.x + threadIdx.x;
  if (idx >= total) return;
  float v = acc[idx] / z[idx >> 6] + b[idx & (HDIM - 1)];
  if (do_relu) v = fmaxf(v, 0.f);
  out[idx] = v;
}

extern "C" void kernel_launch(void* const* d_in, const int* in_sizes, int n_in,
                              void* d_out, int out_size, void* d_ws, size_t ws_size,
                              hipStream_t stream) {
  (void)n_in; (void)out_size; (void)ws_size;
  const int*   ei  = (const int*)d_in[0];
  const float* stu = (const float*)d_in[1];
  const float* crs = (const float*)d_in[2];
  const float* W1  = (const float*)d_in[3];
  const float* a1s = (const float*)d_in[4];
  const float* a1d = (const float*)d_in[5];
  const float* b1  = (const float*)d_in[6];
  const float* W2  = (const float*)d_in[7];
  const float* a2s = (const float*)d_in[8];
  const float* a2d = (const float*)d_in[9];
  const float* b2  = (const float*)d_in[10];

  const int E  = in_sizes[0] / 2;
  const int NS = in_sizes[1] / HDIM;
  const int NC = in_sizes[2] / HDIM;
  const int N  = NS + NC;
  const long NH = (long)N * HDIM;
  const int EN = E + N;

  // workspace carve-up: xbuf | h | s | d | z | m   (~63 MB)
  char* ws = (char*)d_ws;
  float* xbuf = (float*)ws; ws += NH * sizeof(float);
  float* hbuf = (float*)ws; ws += NH * sizeof(float);
  float* sbuf = (float*)ws; ws += (long)N * sizeof(float);
  float* dbuf = (float*)ws; ws += (long)N * sizeof(float);
  float* zbuf = (float*)ws; ws += (long)N * sizeof(float);
  int*   mbuf = (int*)ws;
  float* accb = (float*)d_out;   // d_out doubles as the segment-sum accumulator
  float* outp = (float*)d_out;

  // x = concat(student_embed, course_embed)
  hipMemcpyAsync(xbuf, stu, (size_t)NS * HDIM * sizeof(float),
                 hipMemcpyDeviceToDevice, stream);
  hipMemcpyAsync(xbuf + (long)NS * HDIM, crs, (size_t)NC * HDIM * sizeof(float),
                 hipMemcpyDeviceToDevice, stream);

  const int TB = 256;                       // 8 wave32s per block
  dim3 projGrid((N / 16 + 7) / 8);
  dim3 scoreGrid((int)(((long)N * 32 + TB - 1) / TB));
  dim3 nodeGrid((int)((NH + TB - 1) / TB));
  dim3 edgeGrid((EN + TB - 1) / TB);
  dim3 edgeWGrid((int)(((long)EN * 32 + TB - 1) / TB));

  // ---------------- layer 1 ----------------
  gat_proj_wmma<<<projGrid, TB, 0, stream>>>(xbuf, W1, hbuf, N);
  gat_scores<<<scoreGrid, TB, 0, stream>>>(hbuf, a1s, a1d, sbuf, dbuf, N);
  gat_init<<<nodeGrid, TB, 0, stream>>>(accb, zbuf, mbuf, (int)NH);
  gat_edge_max<<<edgeGrid, TB, 0, stream>>>(ei, E, N, sbuf, dbuf, mbuf);
  gat_edge_accum<<<edgeWGrid, TB, 0, stream>>>(ei, E, N, sbuf, dbuf, mbuf, hbuf, zbuf, accb);
  gat_finalize<<<nodeGrid, TB, 0, stream>>>(accb, zbuf, b1, xbuf, (int)NH, 1); // ReLU -> x2

  // ---------------- layer 2 ----------------
  gat_proj_wmma<<<projGrid, TB, 0, stream>>>(xbuf, W2, hbuf, N);
  gat_scores<<<scoreGrid, TB, 0, stream>>>(hbuf, a2s, a2d, sbuf, dbuf, N);
  gat_init<<<nodeGrid, TB, 0, stream>>>(accb, zbuf, mbuf, (int)NH);
  gat_edge_max<<<edgeGrid, TB, 0, stream>>>(ei, E, N, sbuf, dbuf, mbuf);
  gat_edge_accum<<<edgeWGrid, TB, 0, stream>>>(ei, E, N, sbuf, dbuf, mbuf, hbuf, zbuf, accb);
  gat_finalize<<<nodeGrid, TB, 0, stream>>>(accb, zbuf, b2, outp, (int)NH, 0); // in-place
}